// AttentionHexMeshQNet_50345606644283
// MI455X (gfx1250) — compile-verified
//
#include <hip/hip_runtime.h>
#include <hip/hip_bf16.h>
#include <math.h>

// Problem constants (from reference)
#define Nn   30000
#define Fdim 16
#define Ed   300000
#define HDc  256
#define NHh  8
#define Ch   32    // HD/NH
#define Lg   3
#define Ss   256
#define Kk   128
#define NHCA 4
#define HDCA 64    // HD/NH_CA
#define EN   (Ed + Nn)   // edges incl. self loops

typedef float v2f __attribute__((ext_vector_type(2)));
typedef float v8f __attribute__((ext_vector_type(8)));

// ---------- order-preserving float <-> uint for atomic segment-max ----------
__device__ __forceinline__ unsigned fenc(float x) {
    unsigned u = __float_as_uint(x);
    return (u & 0x80000000u) ? ~u : (u | 0x80000000u);
}
__device__ __forceinline__ float fdec(unsigned e) {
    unsigned u = (e & 0x80000000u) ? (e ^ 0x80000000u) : ~e;
    return __uint_as_float(u);
}

// ---------------- fill helpers ----------------
__global__ void fill_f32(float* p, float v, long n) {
    long i = (long)blockIdx.x * 256 + threadIdx.x;
    if (i < n) p[i] = v;
}
__global__ void fill_u32(unsigned* p, unsigned v, long n) {
    long i = (long)blockIdx.x * 256 + threadIdx.x;
    if (i < n) p[i] = v;
}

// ---------------- pre-transform: h = x @ pre_W + pre_b ----------------
__global__ void pre_kernel(const float* __restrict__ x, const float* __restrict__ W,
                           const float* __restrict__ b, float* __restrict__ h) {
    int n = blockIdx.x, c = threadIdx.x;
    float v = b[c];
    #pragma unroll
    for (int f = 0; f < Fdim; ++f) v += x[n * Fdim + f] * W[f * HDc + c];
    h[(size_t)n * HDc + c] = v;
}

// ---------------- fp32 WMMA GEMM: Y[M,Ncols] = A[M,K] @ W[K,Ncols] (+bias) ----
// 4 waves per block; each wave computes one 16x16 output tile with
// V_WMMA_F32_16X16X4_F32 over K/4 steps. Requires M%16==0, Ncols%64==0 or 128,
// K%16==0 (true for all launches here). No divergent control flow: EXEC stays
// all-ones around every WMMA (ISA 7.12 requirement).
__global__ void gemm_wmma_f32(const float* __restrict__ A, const float* __restrict__ W,
                              const float* __restrict__ bias, float* __restrict__ Y,
                              int K, int Ncols) {
    int lane = threadIdx.x & 31;
    int wave = threadIdx.x >> 5;                 // 0..3
    int ct   = blockIdx.x * 4 + wave;            // column tile
    int rt   = blockIdx.y;                       // row tile
    int half = lane >> 4;                        // 0 or 1
    int l15  = lane & 15;
    int row  = rt * 16 + l15;

    // A fragment (16x4 f32): lane l -> A[l&15][k0 + 2*(l>>4) + {0,1}] (contiguous pair)
    const float* Ap = A + (size_t)row * K + 2 * half;
    // B fragment (4x16 f32): lane l -> W[k0 + 2*(l>>4) + {0,1}][ct*16 + (l&15)]
    const float* Bp = W + (size_t)(2 * half) * Ncols + ct * 16 + l15;

    v8f acc = {};
    #pragma unroll 4
    for (int k0 = 0; k0 < K; k0 += 4) {
        v2f a = *(const v2f*)Ap;                 // 8B-aligned pair load
        v2f b;
        b.x = Bp[0];
        b.y = Bp[Ncols];
        acc = __builtin_amdgcn_wmma_f32_16x16x4_f32(
                  false, a, false, b, (short)0, acc, false, false);
        Ap += 4;
        Bp += (size_t)4 * Ncols;
    }

    // C/D layout: VGPR r: lanes0-15 -> M=rt*16+r, lanes16-31 -> M=rt*16+r+8, N=l15
    float bv = bias ? bias[ct * 16 + l15] : 0.0f;
    float* Yp = Y + (size_t)(rt * 16 + 8 * half) * Ncols + ct * 16 + l15;
    #pragma unroll
    for (int r = 0; r < 8; ++r)
        Yp[(size_t)r * Ncols] = acc[r] + bv;
}

// ---------------- edge kernels (GATv2 attention) ----------------
// e[edge,h] = sum_c lrelu(xl[src]+xr[dst]) * att[h,c]; atomicMax segment max.
__global__ void edge_e_kernel(const float* __restrict__ xl, const float* __restrict__ xr,
                              const float* __restrict__ attw, const int* __restrict__ ei,
                              float* __restrict__ ebuf, unsigned* __restrict__ emax) {
    int e = blockIdx.x;
    int t = threadIdx.x;
    int h = t >> 5, lane = t & 31;
    int s, d;
    if (e < Ed) { s = ei[e]; d = ei[Ed + e]; } else { s = e - Ed; d = s; }
    float m = xl[(size_t)s * HDc + h * Ch + lane] + xr[(size_t)d * HDc + h * Ch + lane];
    m = (m > 0.0f) ? m : 0.2f * m;                       // leaky_relu(0.2)
    float p = m * attw[h * Ch + lane];
    #pragma unroll
    for (int o = 16; o > 0; o >>= 1) p += __shfl_xor(p, o, 32);
    if (lane == 0) {
        ebuf[(size_t)e * NHh + h] = p;
        atomicMax(&emax[(size_t)d * NHh + h], fenc(p));
    }
}

// a = exp(e - emax[dst]); denom[dst] += a
__global__ void edge_a_kernel(float* __restrict__ ebuf, const unsigned* __restrict__ emax,
                              const int* __restrict__ ei, float* __restrict__ denom) {
    long idx = (long)blockIdx.x * 256 + threadIdx.x;
    if (idx >= (long)EN * NHh) return;
    int e = (int)(idx / NHh), h = (int)(idx % NHh);
    int d = (e < Ed) ? ei[Ed + e] : (e - Ed);
    float a = expf(ebuf[idx] - fdec(emax[(size_t)d * NHh + h]));
    ebuf[idx] = a;
    atomicAdd(&denom[(size_t)d * NHh + h], a);
}

// h_new[dst] += (a/denom[dst]) * xl[src]
__global__ void edge_agg_kernel(const float* __restrict__ abuf, const float* __restrict__ denom,
                                const float* __restrict__ xl, const int* __restrict__ ei,
                                float* __restrict__ hnew) {
    int e = blockIdx.x, t = threadIdx.x, h = t >> 5;
    int s, d;
    if (e < Ed) { s = ei[e]; d = ei[Ed + e]; } else { s = e - Ed; d = s; }
    float alpha = abuf[(size_t)e * NHh + h] / (denom[(size_t)d * NHh + h] + 1e-16f);
    atomicAdd(&hnew[(size_t)d * HDc + t], alpha * xl[(size_t)s * HDc + t]);
}

// h = elu(h_new + bias)
__global__ void node_update_kernel(const float* __restrict__ hnew,
                                   const float* __restrict__ bias, float* __restrict__ h) {
    size_t idx = (size_t)blockIdx.x * 256 + threadIdx.x;
    float v = hnew[idx] + bias[threadIdx.x];
    h[idx] = (v > 0.0f) ? v : expm1f(v);
}

// ---------------- pooling / gathers ----------------
__global__ void pool_kernel(const float* __restrict__ h, float* __restrict__ g) {
    int t = threadIdx.x;
    int n0 = blockIdx.x * 256;
    float s = 0.0f;
    for (int i = 0; i < 256; ++i) {
        int n = n0 + i;
        if (n < Nn) s += h[(size_t)n * HDc + t];
    }
    atomicAdd(&g[t], s);
}
__global__ void scale_kernel(float* g, float sc) { g[threadIdx.x] *= sc; }

__global__ void sheet_gather_kernel(const float* __restrict__ h, const int* __restrict__ idx,
                                    float* __restrict__ sheet) {
    int s = blockIdx.x, t = threadIdx.x;
    float acc = 0.0f;
    for (int k2 = 0; k2 < Kk; ++k2)
        acc += h[(size_t)idx[s * Kk + k2] * HDc + t];
    sheet[(size_t)s * HDc + t] = acc * (1.0f / Kk);
}

// ---------------- geo encoder layer 1 (16->256 matmul fused with LN+relu) ----
__global__ void geo1_kernel(const float* __restrict__ sf, const float* __restrict__ W1,
                            const float* __restrict__ b1, const float* __restrict__ g1,
                            const float* __restrict__ be1, float* __restrict__ out) {
    int r = blockIdx.x, t = threadIdx.x;
    float v = b1[t];
    #pragma unroll
    for (int f = 0; f < Fdim; ++f) v += sf[r * Fdim + f] * W1[f * HDc + t];
    __shared__ float sb[256];
    sb[t] = v; __syncthreads();
    for (int o = 128; o > 0; o >>= 1) { if (t < o) sb[t] += sb[t + o]; __syncthreads(); }
    float mu = sb[0] / 256.0f; __syncthreads();
    float dd = v - mu;
    sb[t] = dd * dd; __syncthreads();
    for (int o = 128; o > 0; o >>= 1) { if (t < o) sb[t] += sb[t + o]; __syncthreads(); }
    float var = sb[0] / 256.0f;
    float y = dd * rsqrtf(var + 1e-5f) * g1[t] + be1[t];
    out[r * HDc + t] = fmaxf(y, 0.0f);
}

// ---------------- generic LayerNorm + ReLU (blockDim == cols) ----------------
__global__ void ln_relu_kernel(const float* __restrict__ in, float* __restrict__ out,
                               const float* __restrict__ gamma, const float* __restrict__ beta,
                               int cols) {
    int r = blockIdx.x, t = threadIdx.x;
    float v = in[(size_t)r * cols + t];
    __shared__ float sb[512];
    sb[t] = v; __syncthreads();
    for (int o = cols >> 1; o > 0; o >>= 1) { if (t < o) sb[t] += sb[t + o]; __syncthreads(); }
    float mu = sb[0] / cols; __syncthreads();
    float dd = v - mu;
    sb[t] = dd * dd; __syncthreads();
    for (int o = cols >> 1; o > 0; o >>= 1) { if (t < o) sb[t] += sb[t + o]; __syncthreads(); }
    float var = sb[0] / cols;
    float y = dd * rsqrtf(var + 1e-5f) * gamma[t] + beta[t];
    out[(size_t)r * cols + t] = fmaxf(y, 0.0f);
}

// ---------------- cross attention ----------------
__global__ void attn_kernel(const float* __restrict__ q, const float* __restrict__ k,
                            float* __restrict__ att) {
    int qr = blockIdx.x, h = blockIdx.y, t = threadIdx.x;
    float s = 0.0f;
    #pragma unroll 8
    for (int d2 = 0; d2 < HDCA; ++d2)
        s += q[(size_t)qr * HDc + h * HDCA + d2] * k[(size_t)t * HDc + h * HDCA + d2];
    s *= 0.125f;  // 1/sqrt(64)
    __shared__ float sb[256];
    sb[t] = s; __syncthreads();
    for (int o = 128; o > 0; o >>= 1) { if (t < o) sb[t] = fmaxf(sb[t], sb[t + o]); __syncthreads(); }
    float mx = sb[0]; __syncthreads();
    float ex = expf(s - mx);
    sb[t] = ex; __syncthreads();
    for (int o = 128; o > 0; o >>= 1) { if (t < o) sb[t] += sb[t + o]; __syncthreads(); }
    att[((size_t)h * Ss + qr) * Ss + t] = ex / sb[0];
}

__global__ void attn_apply_kernel(const float* __restrict__ att, const float* __restrict__ v,
                                  float* __restrict__ fused) {
    int qr = blockIdx.x, t = threadIdx.x;
    int h = t >> 6, d2 = t & 63;
    float acc = 0.0f;
    for (int kk = 0; kk < Ss; ++kk)
        acc += att[((size_t)h * Ss + qr) * Ss + kk] * v[(size_t)kk * HDc + h * HDCA + d2];
    fused[(size_t)qr * HDc + t] = acc;
}

// ---------------- concat [fused | g] ----------------
__global__ void concat_kernel(const float* __restrict__ ca, const float* __restrict__ g,
                              float* __restrict__ hq) {
    int s = blockIdx.x, t = threadIdx.x;  // 512 threads
    hq[(size_t)s * (2 * HDc) + t] = (t < HDc) ? ca[(size_t)s * HDc + t] : g[t - HDc];
}

// ---------------- final heads ----------------
__global__ void final_q_kernel(const float* __restrict__ t3, const float* __restrict__ W3,
                               const float* __restrict__ b3, float* __restrict__ out) {
    int s = threadIdx.x;  // 256
    float acc = b3[0];
    for (int j = 0; j < 128; ++j) acc += t3[s * 128 + j] * W3[j];
    out[s] = acc;
}

__global__ void done_kernel(const float* __restrict__ g, const float* __restrict__ W1,
                            const float* __restrict__ b1, const float* __restrict__ ga,
                            const float* __restrict__ be, const float* __restrict__ W2,
                            const float* __restrict__ b2, float* __restrict__ out) {
    int j = threadIdx.x;  // 128
    float v = b1[j];
    for (int c = 0; c < HDc; ++c) v += g[c] * W1[c * 128 + j];
    __shared__ float sb[128];
    sb[j] = v; __syncthreads();
    for (int o = 64; o > 0; o >>= 1) { if (j < o) sb[j] += sb[j + o]; __syncthreads(); }
    float mu = sb[0] / 128.0f; __syncthreads();
    float dd = v - mu;
    sb[j] = dd * dd; __syncthreads();
    for (int o = 64; o > 0; o >>= 1) { if (j < o) sb[j] += sb[j + o]; __syncthreads(); }
    float var = sb[0] / 128.0f; __syncthreads();
    float y = fmaxf(dd * rsqrtf(var + 1e-5f) * ga[j] + be[j], 0.0f);
    sb[j] = y * W2[j]; __syncthreads();
    for (int o = 64; o > 0; o >>= 1) { if (j < o) sb[j] += sb[j + o]; __syncthreads(); }
    if (j == 0) out[0] = sb[0] + b2[0];
}

// =============================== host launch ===============================
extern "C" void kernel_launch(void* const* d_in, const int* in_sizes, int n_in,
                              void* d_out, int out_size, void* d_ws, size_t ws_size,
                              hipStream_t stream) {
    const float* x      = (const float*)d_in[0];
    const int*   ei     = (const int*)  d_in[1];
    // d_in[2] = batch (all zeros, unused)
    const int*   sidx   = (const int*)  d_in[3];
    const float* sf     = (const float*)d_in[4];
    const float* pre_W  = (const float*)d_in[5];
    const float* pre_b  = (const float*)d_in[6];
    const float* gWl    = (const float*)d_in[7];
    const float* gWr    = (const float*)d_in[8];
    const float* gatt   = (const float*)d_in[9];
    const float* gb     = (const float*)d_in[10];
    const float* geoW1  = (const float*)d_in[11];
    const float* geob1  = (const float*)d_in[12];
    const float* geog1  = (const float*)d_in[13];
    const float* geobe1 = (const float*)d_in[14];
    const float* geoW2  = (const float*)d_in[15];
    const float* geob2  = (const float*)d_in[16];
    const float* geog2  = (const float*)d_in[17];
    const float* geobe2 = (const float*)d_in[18];
    const float* caWq   = (const float*)d_in[19];
    const float* cabq   = (const float*)d_in[20];
    const float* caWk   = (const float*)d_in[21];
    const float* cabk   = (const float*)d_in[22];
    const float* caWv   = (const float*)d_in[23];
    const float* cabv   = (const float*)d_in[24];
    const float* caWo   = (const float*)d_in[25];
    const float* cabo   = (const float*)d_in[26];
    const float* qW1    = (const float*)d_in[27];
    const float* qb1    = (const float*)d_in[28];
    const float* qg1    = (const float*)d_in[29];
    const float* qbe1   = (const float*)d_in[30];
    const float* qW2    = (const float*)d_in[31];
    const float* qb2    = (const float*)d_in[32];
    const float* qg2    = (const float*)d_in[33];
    const float* qbe2   = (const float*)d_in[34];
    const float* qW3    = (const float*)d_in[35];
    const float* qb3    = (const float*)d_in[36];
    const float* dW1    = (const float*)d_in[37];
    const float* db1    = (const float*)d_in[38];
    const float* dg1    = (const float*)d_in[39];
    const float* dbe1   = (const float*)d_in[40];
    const float* dW2    = (const float*)d_in[41];
    const float* db2    = (const float*)d_in[42];

    float* ws = (float*)d_ws;
    // workspace layout (floats)
    size_t o_h     = 0;
    size_t o_xl    = o_h     + (size_t)Nn * HDc;
    size_t o_xr    = o_xl    + (size_t)Nn * HDc;
    size_t o_hnew  = o_xr    + (size_t)Nn * HDc;
    size_t o_e     = o_hnew  + (size_t)Nn * HDc;       // EN*8 (reused for a)
    size_t o_emax  = o_e     + (size_t)EN * NHh;       // uint-encoded
    size_t o_denom = o_emax  + (size_t)Nn * NHh;
    size_t o_g     = o_denom + (size_t)Nn * NHh;       // 256
    size_t o_sheet = o_g     + 256;
    size_t o_geo   = o_sheet + (size_t)Ss * HDc;
    size_t o_t1    = o_geo   + (size_t)Ss * HDc;       // 256x256 temp
    size_t o_q     = o_t1    + (size_t)Ss * HDc;
    size_t o_k     = o_q     + (size_t)Ss * HDc;
    size_t o_v     = o_k     + (size_t)Ss * HDc;
    size_t o_att   = o_v     + (size_t)Ss * HDc;       // 4*256*256
    size_t o_fused = o_att   + (size_t)NHCA * Ss * Ss;
    size_t o_ca    = o_fused + (size_t)Ss * HDc;
    size_t o_hq    = o_ca    + (size_t)Ss * HDc;       // 256x512
    size_t o_t2    = o_hq    + (size_t)Ss * 2 * HDc;   // 256x256
    size_t o_t3    = o_t2    + (size_t)Ss * HDc;       // 256x128

    float*    h     = ws + o_h;
    float*    xl    = ws + o_xl;
    float*    xr    = ws + o_xr;
    float*    hnew  = ws + o_hnew;
    float*    ebuf  = ws + o_e;
    unsigned* emax  = (unsigned*)(ws + o_emax);
    float*    denom = ws + o_denom;
    float*    g     = ws + o_g;
    float*    sheet = ws + o_sheet;
    float*    geo   = ws + o_geo;
    float*    t1    = ws + o_t1;
    float*    qb_   = ws + o_q;
    float*    kb    = ws + o_k;
    float*    vb    = ws + o_v;
    float*    att   = ws + o_att;
    float*    fused = ws + o_fused;
    float*    ca    = ws + o_ca;
    float*    hq    = ws + o_hq;
    float*    t2    = ws + o_t2;
    float*    t3    = ws + o_t3;

    float* outp = (float*)d_out;  // [0..255] q_values, [256] done_logit

    // 1) pre-transform
    pre_kernel<<<Nn, HDc, 0, stream>>>(x, pre_W, pre_b, h);

    // 2) GATv2 layers
    for (int l = 0; l < Lg; ++l) {
        gemm_wmma_f32<<<dim3(HDc / 64, Nn / 16), 128, 0, stream>>>(
            h, gWl + (size_t)l * HDc * HDc, nullptr, xl, HDc, HDc);
        gemm_wmma_f32<<<dim3(HDc / 64, Nn / 16), 128, 0, stream>>>(
            h, gWr + (size_t)l * HDc * HDc, nullptr, xr, HDc, HDc);

        long nEM = (long)Nn * NHh;
        fill_u32<<<(unsigned)((nEM + 255) / 256), 256, 0, stream>>>(emax, 0u, nEM);
        fill_f32<<<(unsigned)((nEM + 255) / 256), 256, 0, stream>>>(denom, 0.0f, nEM);
        long nH = (long)Nn * HDc;
        fill_f32<<<(unsigned)((nH + 255) / 256), 256, 0, stream>>>(hnew, 0.0f, nH);

        edge_e_kernel<<<EN, 256, 0, stream>>>(xl, xr, gatt + (size_t)l * NHh * Ch, ei, ebuf, emax);
        long nEA = (long)EN * NHh;
        edge_a_kernel<<<(unsigned)((nEA + 255) / 256), 256, 0, stream>>>(ebuf, emax, ei, denom);
        edge_agg_kernel<<<EN, 256, 0, stream>>>(ebuf, denom, xl, ei, hnew);
        node_update_kernel<<<Nn, HDc, 0, stream>>>(hnew, gb + (size_t)l * HDc, h);
    }

    // 3) global mean pool
    fill_f32<<<1, 256, 0, stream>>>(g, 0.0f, 256);
    pool_kernel<<<(Nn + 255) / 256, 256, 0, stream>>>(h, g);
    scale_kernel<<<1, 256, 0, stream>>>(g, 1.0f / (float)Nn);

    // 4) sheet gather + mean
    sheet_gather_kernel<<<Ss, HDc, 0, stream>>>(h, sidx, sheet);

    // 5) geometry encoder
    geo1_kernel<<<Ss, HDc, 0, stream>>>(sf, geoW1, geob1, geog1, geobe1, geo);
    gemm_wmma_f32<<<dim3(HDc / 64, Ss / 16), 128, 0, stream>>>(geo, geoW2, geob2, t1, HDc, HDc);
    ln_relu_kernel<<<Ss, HDc, 0, stream>>>(t1, geo, geog2, geobe2, HDc);

    // 6) cross attention
    gemm_wmma_f32<<<dim3(HDc / 64, Ss / 16), 128, 0, stream>>>(sheet, caWq, cabq, qb_, HDc, HDc);
    gemm_wmma_f32<<<dim3(HDc / 64, Ss / 16), 128, 0, stream>>>(geo,   caWk, cabk, kb,  HDc, HDc);
    gemm_wmma_f32<<<dim3(HDc / 64, Ss / 16), 128, 0, stream>>>(geo,   caWv, cabv, vb,  HDc, HDc);
    attn_kernel<<<dim3(Ss, NHCA), Ss, 0, stream>>>(qb_, kb, att);
    attn_apply_kernel<<<Ss, HDc, 0, stream>>>(att, vb, fused);
    gemm_wmma_f32<<<dim3(HDc / 64, Ss / 16), 128, 0, stream>>>(fused, caWo, cabo, ca, HDc, HDc);

    // 7) q-MLP
    concat_kernel<<<Ss, 2 * HDc, 0, stream>>>(ca, g, hq);
    gemm_wmma_f32<<<dim3(HDc / 64, Ss / 16), 128, 0, stream>>>(hq, qW1, qb1, t2, 2 * HDc, HDc);
    ln_relu_kernel<<<Ss, HDc, 0, stream>>>(t2, t2, qg1, qbe1, HDc);
    gemm_wmma_f32<<<dim3(128 / 64, Ss / 16), 128, 0, stream>>>(t2, qW2, qb2, t3, HDc, 128);
    ln_relu_kernel<<<Ss, 128, 0, stream>>>(t3, t3, qg2, qbe2, 128);
    final_q_kernel<<<1, Ss, 0, stream>>>(t3, qW3, qb3, outp);

    // 8) done predictor
    done_kernel<<<1, 128, 0, stream>>>(g, dW1, db1, dg1, dbe1, dW2, db2, outp + Ss);

    (void)in_sizes; (void)n_in; (void)out_size; (void)ws_size;
}